// DecompGridv7_78099685310709
// MI455X (gfx1250) — compile-verified
//
#include <hip/hip_runtime.h>
#include <cstddef>

// ---------------------------------------------------------------------------
// Tri-plane (3-resolution) feature query.
//   out[b, lvl*32 + c] = prod over 3 planes of bilerp(plane[lvl][i], coords)
// Strategy:
//   1) transpose each plane (C,H,W) -> (H,W,C) into workspace so one texel's
//      32 channels = one 128B cache line (one L2 request per corner gather).
//   2) query kernel: 8 lanes/point, float4 per lane (wave32 = 4 points/wave),
//      coords staged to LDS via gfx1250 async global->LDS loads,
//      non-temporal 128-bit output stores (planes stay L2-resident).
// ---------------------------------------------------------------------------

typedef __attribute__((ext_vector_type(4))) float v4f;

static constexpr int NF = 32;

// float offsets of the 9 transposed planes inside the workspace
static constexpr size_t PLANE_OFF_F[9] = {
    0u,        131072u,  262144u,          // low  (64x64x32   each)
    393216u,   917504u,  1441792u,         // mid  (128x128x32 each)
    1966080u,  4063232u, 6160384u          // high (256x256x32 each)
};
static constexpr size_t WS_FLOATS = 8257536u;   // 33,030,144 bytes total

#if defined(__AMDGCN__) && __has_builtin(__builtin_amdgcn_global_load_async_to_lds_b32)
#define TPQ_HAVE_ASYNC_LDS 1
typedef __attribute__((address_space(1))) int as1_int;
typedef __attribute__((address_space(3))) int as3_int;
#endif

// ---------------------------------------------------------------------------
// (C,H,W) -> (H,W,C) transpose, 32x32 texel-channel tile through padded LDS.
// Both the global read and the global write are fully coalesced.
// ---------------------------------------------------------------------------
__global__ __launch_bounds__(256) void transpose_plane(
    const float* __restrict__ src, float* __restrict__ dst, int S)
{
    __shared__ float tile[32][33];
    const int tx = threadIdx.x & 31;   // fast index
    const int tz = threadIdx.x >> 5;   // 0..7
    const int tilesPerRow = S >> 5;
    const int y  = blockIdx.x / tilesPerRow;
    const int xb = (blockIdx.x - y * tilesPerRow) << 5;
    const int HW = S * S;

#pragma unroll
    for (int i = 0; i < 4; ++i) {
        const int c = tz + (i << 3);                 // channel 0..31
        tile[c][tx] = src[(size_t)c * HW + (size_t)y * S + xb + tx];
    }
    __syncthreads();
#pragma unroll
    for (int i = 0; i < 4; ++i) {
        const int xp = tz + (i << 3);                // x within tile
        dst[((size_t)(y * S + xb + xp) << 5) + tx] = tile[tx][xp];
    }
}

// ---------------------------------------------------------------------------
// Bilinear sample of 4 consecutive channels from a (H,W,C=32) plane.
// Matches the reference clamp semantics (align_corners=True, x in [-1,1]).
// ---------------------------------------------------------------------------
template <int S>
__device__ __forceinline__ v4f bilerp4(const float* __restrict__ p,
                                       float gx, float gy, int coff)
{
    const float px = (gx + 1.0f) * (0.5f * (float)(S - 1));
    const float py = (gy + 1.0f) * (0.5f * (float)(S - 1));
    const float fx = floorf(px), fy = floorf(py);
    const float wx = px - fx,    wy = py - fy;

    int x0 = (int)fx; x0 = x0 < 0 ? 0 : (x0 > S - 1 ? S - 1 : x0);
    int y0 = (int)fy; y0 = y0 < 0 ? 0 : (y0 > S - 1 ? S - 1 : y0);
    const int x1 = (x0 + 1 > S - 1) ? S - 1 : x0 + 1;
    const int y1 = (y0 + 1 > S - 1) ? S - 1 : y0 + 1;

    const int r0 = y0 * S, r1 = y1 * S;
    const v4f v00 = *(const v4f*)(p + (((size_t)(r0 + x0)) << 5) + coff);
    const v4f v01 = *(const v4f*)(p + (((size_t)(r0 + x1)) << 5) + coff);
    const v4f v10 = *(const v4f*)(p + (((size_t)(r1 + x0)) << 5) + coff);
    const v4f v11 = *(const v4f*)(p + (((size_t)(r1 + x1)) << 5) + coff);

    const float iwx = 1.0f - wx, iwy = 1.0f - wy;
    const float w00 = iwx * iwy, w01 = wx * iwy;
    const float w10 = iwx * wy,  w11 = wx * wy;
    return v00 * w00 + v01 * w01 + v10 * w10 + v11 * w11;
}

// ---------------------------------------------------------------------------
// Main query kernel: 256 threads = 32 points/block; 8 lanes/point (c4 = 0..7),
// each lane handles 4 channels. One wave32 = 4 points.
// ---------------------------------------------------------------------------
__global__ __launch_bounds__(256) void tq_kernel(
    const float* __restrict__ x, const float* __restrict__ T,
    float* __restrict__ out, int batch)
{
    __shared__ float sx[96];                       // 32 points * 3 coords
    const int lt        = threadIdx.x;
    const int blockBase = blockIdx.x << 5;         // first point of block
    const long long base3 = (long long)blockBase * 3;
    const long long lim3  = (long long)batch * 3;

    if (lt < 96 && base3 + lt < lim3) {
#ifdef TPQ_HAVE_ASYNC_LDS
        __builtin_amdgcn_global_load_async_to_lds_b32(
            (as1_int*)(x + base3 + lt),
            (as3_int*)&sx[lt],
            /*offset=*/0, /*cpol=*/0);
#else
        sx[lt] = x[base3 + lt];
#endif
    }
#ifdef TPQ_HAVE_ASYNC_LDS
#if __has_builtin(__builtin_amdgcn_s_wait_asynccnt)
    __builtin_amdgcn_s_wait_asynccnt(0);
#else
    asm volatile("s_wait_asynccnt 0" ::: "memory");
#endif
#endif
    __syncthreads();

    const int pL = lt >> 3;                        // point within block
    const int c4 = lt & 7;                         // channel quad
    const int b  = blockBase + pL;
    if (b >= batch) return;

    const float g0 = sx[pL * 3 + 0];
    const float g1 = sx[pL * 3 + 1];
    const float g2 = sx[pL * 3 + 2];
    const int coff = c4 << 2;
    float* const ob = out + (size_t)b * 96 + coff;

    {   // low (64)
        v4f a = bilerp4<64>(T + PLANE_OFF_F[0], g0, g1, coff);
        a = a * bilerp4<64>(T + PLANE_OFF_F[1], g0, g2, coff);
        a = a * bilerp4<64>(T + PLANE_OFF_F[2], g1, g2, coff);
        __builtin_nontemporal_store(a, (v4f*)(ob + 0));
    }
    {   // mid (128)
        v4f a = bilerp4<128>(T + PLANE_OFF_F[3], g0, g1, coff);
        a = a * bilerp4<128>(T + PLANE_OFF_F[4], g0, g2, coff);
        a = a * bilerp4<128>(T + PLANE_OFF_F[5], g1, g2, coff);
        __builtin_nontemporal_store(a, (v4f*)(ob + 32));
    }
    {   // high (256)
        v4f a = bilerp4<256>(T + PLANE_OFF_F[6], g0, g1, coff);
        a = a * bilerp4<256>(T + PLANE_OFF_F[7], g0, g2, coff);
        a = a * bilerp4<256>(T + PLANE_OFF_F[8], g1, g2, coff);
        __builtin_nontemporal_store(a, (v4f*)(ob + 64));
    }
}

// ---------------------------------------------------------------------------
// Fallback (workspace too small): gather straight from (C,H,W) layout,
// one lane per (point, channel).
// ---------------------------------------------------------------------------
template <int S>
__device__ __forceinline__ float bilerp1(const float* __restrict__ p,
                                         float gx, float gy, int c)
{
    const float px = (gx + 1.0f) * (0.5f * (float)(S - 1));
    const float py = (gy + 1.0f) * (0.5f * (float)(S - 1));
    const float fx = floorf(px), fy = floorf(py);
    const float wx = px - fx,    wy = py - fy;
    int x0 = (int)fx; x0 = x0 < 0 ? 0 : (x0 > S - 1 ? S - 1 : x0);
    int y0 = (int)fy; y0 = y0 < 0 ? 0 : (y0 > S - 1 ? S - 1 : y0);
    const int x1 = (x0 + 1 > S - 1) ? S - 1 : x0 + 1;
    const int y1 = (y0 + 1 > S - 1) ? S - 1 : y0 + 1;
    const float* pc = p + (size_t)c * (S * S);
    const float v00 = pc[y0 * S + x0], v01 = pc[y0 * S + x1];
    const float v10 = pc[y1 * S + x0], v11 = pc[y1 * S + x1];
    const float iwx = 1.0f - wx, iwy = 1.0f - wy;
    return v00 * (iwx * iwy) + v01 * (wx * iwy) + v10 * (iwx * wy) + v11 * (wx * wy);
}

__global__ __launch_bounds__(256) void tq_direct_kernel(
    const float* __restrict__ x,
    const float* __restrict__ l0, const float* __restrict__ l1, const float* __restrict__ l2,
    const float* __restrict__ m0, const float* __restrict__ m1, const float* __restrict__ m2,
    const float* __restrict__ h0, const float* __restrict__ h1, const float* __restrict__ h2,
    float* __restrict__ out, int batch)
{
    const int gid = blockIdx.x * 256 + threadIdx.x;
    const int b = gid >> 5, c = gid & 31;
    if (b >= batch) return;
    const float g0 = x[3 * b + 0], g1 = x[3 * b + 1], g2 = x[3 * b + 2];
    float* const ob = out + (size_t)b * 96 + c;
    ob[0]  = bilerp1<64 >(l0, g0, g1, c) * bilerp1<64 >(l1, g0, g2, c) * bilerp1<64 >(l2, g1, g2, c);
    ob[32] = bilerp1<128>(m0, g0, g1, c) * bilerp1<128>(m1, g0, g2, c) * bilerp1<128>(m2, g1, g2, c);
    ob[64] = bilerp1<256>(h0, g0, g1, c) * bilerp1<256>(h1, g0, g2, c) * bilerp1<256>(h2, g1, g2, c);
}

// ---------------------------------------------------------------------------
extern "C" void kernel_launch(void* const* d_in, const int* in_sizes, int n_in,
                              void* d_out, int out_size, void* d_ws, size_t ws_size,
                              hipStream_t stream)
{
    const float* x = (const float*)d_in[0];
    const float* P[9];
    for (int i = 0; i < 9; ++i) P[i] = (const float*)d_in[1 + i];
    float* out = (float*)d_out;
    const int batch = in_sizes[0] / 3;

    if (d_ws && ws_size >= WS_FLOATS * sizeof(float)) {
        float* T = (float*)d_ws;
        static const int S9[9] = {64, 64, 64, 128, 128, 128, 256, 256, 256};
        for (int i = 0; i < 9; ++i) {
            const int S = S9[i];
            transpose_plane<<<dim3(S * (S >> 5)), dim3(256), 0, stream>>>(
                P[i], T + PLANE_OFF_F[i], S);
        }
        const int nblk = (batch + 31) / 32;
        tq_kernel<<<dim3(nblk), dim3(256), 0, stream>>>(x, T, out, batch);
    } else {
        const long long nthreads = (long long)batch * 32;
        const int nblk = (int)((nthreads + 255) / 256);
        tq_direct_kernel<<<dim3(nblk), dim3(256), 0, stream>>>(
            x, P[0], P[1], P[2], P[3], P[4], P[5], P[6], P[7], P[8], out, batch);
    }
}